// RoINetNew_13597866459572
// MI455X (gfx1250) — compile-verified
//
#include <hip/hip_runtime.h>
#include <math.h>

// ---------------- problem constants (match reference) ----------------
#define BIMG   4
#define NPROP  4000
#define CCLS   80
#define NLOGIT 81
#define NCAND  (NPROP * CCLS)        // 320000 candidates per image
#define NROWS  (BIMG * NPROP)        // 16000 proposal rows
#define TOTC   (BIMG * NCAND)        // 1280000 candidates total
#define DET    100
#define SCORE_TH  0.01f
#define NMS_TH    0.5f
#define MIN_AREA  0.1f
#define BBOX_CLIP 4.135166556742356f // log(1000/16)

typedef __attribute__((ext_vector_type(2))) float v2f;
typedef __attribute__((ext_vector_type(8))) float v8f;
typedef __attribute__((ext_vector_type(4))) int   i4;

// ---------------- CDNA5 feature gates (compile-time probed) ----------------
#if defined(__AMDGCN__) && __has_builtin(__builtin_amdgcn_wmma_f32_16x16x4_f32)
#define USE_WMMA 1
#endif
#if defined(__AMDGCN__) && __has_builtin(__builtin_amdgcn_global_load_async_to_lds_b128)
#define USE_ASYNC 1
typedef __attribute__((address_space(1))) i4 gi4;   // global (AS1) int4
typedef __attribute__((address_space(3))) i4 li4;   // LDS    (AS3) int4
#endif

static __device__ __forceinline__ void wait_async0() {
#if defined(__AMDGCN__)
#if __has_builtin(__builtin_amdgcn_s_wait_asynccnt)
  __builtin_amdgcn_s_wait_asynccnt(0);
#else
  asm volatile("s_wait_asynccnt 0" ::: "memory");
#endif
#endif
}

static __device__ __forceinline__ unsigned ballot32(bool p) {
#if defined(__AMDGCN__) && __has_builtin(__builtin_amdgcn_ballot_w32)
  return __builtin_amdgcn_ballot_w32(p);
#else
  return (unsigned)__ballot(p);
#endif
}

// order-preserving float<->uint encoding so atomicMax works for any sign
static __device__ __forceinline__ unsigned fenc(float f) {
  unsigned u = __float_as_uint(f);
  return (u & 0x80000000u) ? ~u : (u | 0x80000000u);
}
static __device__ __forceinline__ float fdec(unsigned e) {
  unsigned u = (e & 0x80000000u) ? (e & 0x7FFFFFFFu) : ~e;
  return __uint_as_float(u);
}

// ---------------- k0: init per-image max + compact counters ----------------
__global__ void init_kernel(unsigned* img_max_enc, int* cnt) {
  int t = threadIdx.x;
  if (t < BIMG) { img_max_enc[t] = fenc(-__builtin_inff()); cnt[t] = 0; }
}

// ---------------- k1: softmax + box decode + validity mask ----------------
// One wave (32 lanes) owns a tile of 16 proposal rows (16x81 logits, 1280
// class-candidates). Logit tile is staged to LDS via async-to-LDS B128
// (ASYNCcnt), row max via 2-lanes-per-row + shfl_xor(16), row sums of
// exp(x-max) via V_WMMA_F32_16X16X4_F32 against an all-ones B matrix
// (D[m][n] = rowsum(m) for every n), then 1280 decodes with coalesced
// float4 delta loads. Tiles never cross an image boundary (4000 % 16 == 0).
__global__ __launch_bounds__(256) void decode_kernel(
    const float* __restrict__ label_pre, const float* __restrict__ bbox_pre,
    const float* __restrict__ proposals, float* __restrict__ boxes_ws,
    float* __restrict__ masked_ws, unsigned* __restrict__ img_max_enc) {
  __shared__ __align__(16) float s_log[8][16 * NLOGIT]; // 8 waves x 1296 floats
  __shared__ float  s_max[8][16];
  __shared__ float  s_den[8][16];
  __shared__ float4 s_prop[8][16];

  const int lane = threadIdx.x & 31;
  const int w    = threadIdx.x >> 5;
  const int tile = blockIdx.x * 8 + w;
  const int r0   = tile * 16;
  const int img  = r0 / NPROP;
  const float* gbase = label_pre + (size_t)r0 * NLOGIT; // 1296 floats, 16B aligned

#if defined(USE_ASYNC)
  // 324 x B128 = 1296 dwords; gfx1250 async global->LDS DMA path
  for (int t = 0; t < 11; ++t) {
    int vi = t * 32 + lane;
    if (vi < 324) {
      __builtin_amdgcn_global_load_async_to_lds_b128(
          (gi4*)(gbase + vi * 4), (li4*)&s_log[w][vi * 4], 0, 0);
    }
  }
  wait_async0();
#else
  for (int i = lane; i < 16 * NLOGIT; i += 32) s_log[w][i] = gbase[i];
#endif
  if (lane < 16) s_prop[w][lane] = ((const float4*)proposals)[r0 + lane];
  __syncthreads();

  // per-row max: lane L and lane 16+L split row L&15
  const int row = lane & 15;
  const int half = lane >> 4;
  float m = -__builtin_inff();
  {
    int c0 = half * 41, c1 = half ? NLOGIT : 41;
    for (int c = c0; c < c1; ++c) m = fmaxf(m, s_log[w][row * NLOGIT + c]);
    m = fmaxf(m, __shfl_xor(m, 16, 32));
  }
  if (lane < 16) s_max[w][row] = m;

#if defined(USE_WMMA)
  { // row sums of exp on the matrix pipe: A(16x84 padded) x ones(84x16)
    v8f acc = {0.f, 0.f, 0.f, 0.f, 0.f, 0.f, 0.f, 0.f};
    v2f ones; ones.x = 1.0f; ones.y = 1.0f;
    for (int j = 0; j < 21; ++j) {
      int c0 = j * 4 + half * 2; // A layout: lanes 0-15 hold K=0,1; 16-31 hold K=2,3
      v2f a;
      a.x = (c0     < NLOGIT) ? __expf(s_log[w][row * NLOGIT + c0]     - m) : 0.0f;
      a.y = (c0 + 1 < NLOGIT) ? __expf(s_log[w][row * NLOGIT + c0 + 1] - m) : 0.0f;
      acc = __builtin_amdgcn_wmma_f32_16x16x4_f32(false, a, false, ones,
                                                  (short)0, acc, false, false);
    }
    // D layout: lanes 0-15 hold M=0..7 in v0..7, lanes 16-31 hold M=8..15
    if (lane == 0)  for (int i = 0; i < 8; ++i) s_den[w][i]     = acc[i];
    if (lane == 16) for (int i = 0; i < 8; ++i) s_den[w][8 + i] = acc[i];
  }
#else
  {
    float ssum = 0.f;
    int c0 = half * 41, c1 = half ? NLOGIT : 41;
    for (int c = c0; c < c1; ++c) ssum += __expf(s_log[w][row * NLOGIT + c] - m);
    ssum += __shfl_xor(ssum, 16, 32);
    if (lane < 16) s_den[w][row] = ssum;
  }
#endif
  __syncthreads();

  // 1280 candidates per wave: consecutive lanes -> consecutive classes ->
  // contiguous float4 delta loads (fully coalesced B128).
  float mx = -__builtin_inff();
  for (int t = lane; t < 16 * CCLS; t += 32) {
    int lr = t / CCLS;
    int c  = t - lr * CCLS;  // label-1
    int gr = r0 + lr;
    float score = __expf(s_log[w][lr * NLOGIT + c + 1] - s_max[w][lr]) / s_den[w][lr];
    float4 dl = *(const float4*)(bbox_pre + (size_t)gr * (CCLS * 4) + c * 4);
    float4 p  = s_prop[w][lr];
    float pw = p.z - p.x, ph = p.w - p.y;
    float cx = p.x + 0.5f * pw, cy = p.y + 0.5f * ph;
    float dw = fminf(dl.z, BBOX_CLIP), dh = fminf(dl.w, BBOX_CLIP);
    float ncx = dl.x * pw + cx, ncy = dl.y * ph + cy;
    float nw = __expf(dw) * pw, nh = __expf(dh) * ph;
    float x1 = ncx - 0.5f * nw, y1 = ncy - 0.5f * nh;
    float x2 = ncx + 0.5f * nw, y2 = ncy + 0.5f * nh;
    float area = (y2 - y1) * (x2 - x1);
    bool valid = (score > SCORE_TH) && (area > MIN_AREA);
    size_t e = (size_t)gr * CCLS + c; // == img*NCAND + (n*C + c)
    ((float4*)boxes_ws)[e] = make_float4(x1, y1, x2, y2);
    masked_ws[e] = valid ? score : -1.0f;
    mx = fmaxf(mx, fmaxf(fmaxf(x1, y1), fmaxf(x2, y2)));
  }
  for (int off = 16; off; off >>= 1) mx = fmaxf(mx, __shfl_xor(mx, off, 32));
  if (lane == 0) atomicMax(&img_max_enc[img], fenc(mx));
}

// ---------------- k2: compact live candidates (wave-aggregated atomics) ----
__global__ void compact_kernel(const float* __restrict__ masked,
                               const float* __restrict__ boxes,
                               const unsigned* __restrict__ img_max_enc,
                               int* __restrict__ cnt, int* __restrict__ cidx,
                               float* __restrict__ cscore, float* __restrict__ cbox) {
  const int lane = threadIdx.x & 31;
  const int stride = gridDim.x * blockDim.x;
  for (int e = blockIdx.x * blockDim.x + threadIdx.x; e < TOTC; e += stride) {
    int img = e / NCAND;        // NCAND % 32 == 0 -> whole wave shares img
    int idx = e - img * NCAND;
    float s = masked[e];
    bool live = s > 0.0f;       // masked is either >SCORE_TH or exactly -1
    unsigned mask = ballot32(live);
    if (mask) {
      int leader = __ffs(mask) - 1;
      int base = 0;
      if (lane == leader) base = atomicAdd(&cnt[img], __popc(mask));
      base = __shfl(base, leader, 32);
      if (live) {
        int ci = img * NCAND + base + __popc(mask & ((1u << lane) - 1u));
        cidx[ci] = idx;
        cscore[ci] = s;
        float off = (float)(idx % CCLS + 1) * (fdec(img_max_enc[img]) + 1.0f);
        float4 b = ((const float4*)boxes)[e];
        ((float4*)cbox)[ci] = make_float4(b.x + off, b.y + off, b.z + off, b.w + off);
      }
    }
  }
}

// ---------------- k3: sequential NMS, one workgroup (32 waves) per image ----
__global__ __launch_bounds__(1024) void nms_kernel(
    const int* __restrict__ cnt, const int* __restrict__ cidx,
    float* __restrict__ cscore, const float* __restrict__ cbox,
    const float* __restrict__ boxes_ws, float* __restrict__ out) {
  const int img = blockIdx.x;
  const int tid = threadIdx.x;
  const int lane = tid & 31, wid = tid >> 5;
  const int n_live = cnt[img];
  const int cb = img * NCAND;
  float* sc = cscore + cb;
  const int* ix = cidx + cb;
  const float4* obox = ((const float4*)cbox) + cb;
  float* ob_out = out;                       // boxes  (B,100,4)
  float* os_out = out + BIMG * DET * 4;      // scores (B,100)
  float* ol_out = os_out + BIMG * DET;       // labels (B,100) as float

  __shared__ float wv[32]; __shared__ int wi[32]; __shared__ int wj[32];
  __shared__ float sel_box[4];
  __shared__ float sel_v; __shared__ int sel_i; __shared__ int sel_j;
  __shared__ int s_done;
  if (tid == 0) s_done = 0;
  __syncthreads();

  for (int d = 0; d < DET; ++d) {
    const int done = s_done;
    // ---- argmax over remaining (tie-break: lowest ORIGINAL index) ----
    float bv = -__builtin_inff(); int bi = 0x7FFFFFFF; int bj = -1;
    if (!done) {
      for (int j = tid; j < n_live; j += 1024) {
        __builtin_prefetch(sc + j + 1024, 0, 1); // global_prefetch_b8
        float s = sc[j]; int id = ix[j];
        if (s > bv || (s == bv && id < bi)) { bv = s; bi = id; bj = j; }
      }
      for (int off = 16; off; off >>= 1) {
        float ov = __shfl_down(bv, off, 32);
        int oi = __shfl_down(bi, off, 32);
        int oj = __shfl_down(bj, off, 32);
        if (ov > bv || (ov == bv && oi < bi)) { bv = ov; bi = oi; bj = oj; }
      }
      if (lane == 0) { wv[wid] = bv; wi[wid] = bi; wj[wid] = bj; }
    }
    __syncthreads();
    if (!done && wid == 0) {
      bv = wv[lane]; bi = wi[lane]; bj = wj[lane];
      for (int off = 16; off; off >>= 1) {
        float ov = __shfl_down(bv, off, 32);
        int oi = __shfl_down(bi, off, 32);
        int oj = __shfl_down(bj, off, 32);
        if (ov > bv || (ov == bv && oi < bi)) { bv = ov; bi = oi; bj = oj; }
      }
      if (lane == 0) { sel_v = bv; sel_i = bi; sel_j = bj; }
    }
    __syncthreads();
    // ---- emit detection d; broadcast selected offset-box ----
    if (tid == 0) {
      float v = done ? -1.0f : sel_v;
      int o = img * DET + d;
      if (v > 0.0f) {
        const float* rb = boxes_ws + (size_t)(cb + sel_i) * 4;
        ob_out[o * 4 + 0] = rb[0]; ob_out[o * 4 + 1] = rb[1];
        ob_out[o * 4 + 2] = rb[2]; ob_out[o * 4 + 3] = rb[3];
        os_out[o] = v;
        ol_out[o] = (float)(sel_i % CCLS + 1);
        float4 sb = obox[sel_j];
        sel_box[0] = sb.x; sel_box[1] = sb.y; sel_box[2] = sb.z; sel_box[3] = sb.w;
        sc[sel_j] = -1.0f;
      } else {
        ob_out[o * 4 + 0] = 0.f; ob_out[o * 4 + 1] = 0.f;
        ob_out[o * 4 + 2] = 0.f; ob_out[o * 4 + 3] = 0.f;
        os_out[o] = 0.f; ol_out[o] = 0.f;
        s_done = 1; // scores only shrink; all later iterations are invalid too
      }
    }
    __syncthreads();
    // ---- suppress IoU > 0.5 against remaining live entries ----
    if (!s_done) {
      float ax1 = sel_box[0], ay1 = sel_box[1], ax2 = sel_box[2], ay2 = sel_box[3];
      float a1 = (ax2 - ax1) * (ay2 - ay1);
      for (int j = tid; j < n_live; j += 1024) {
        __builtin_prefetch(&obox[j + 1024], 0, 1);
        float s = sc[j];
        if (s > 0.0f) {
          float4 b = obox[j];
          float x1 = fmaxf(ax1, b.x), y1 = fmaxf(ay1, b.y);
          float x2 = fminf(ax2, b.z), y2 = fminf(ay2, b.w);
          float inter = fmaxf(x2 - x1, 0.f) * fmaxf(y2 - y1, 0.f);
          float a2 = (b.z - b.x) * (b.w - b.y);
          float iou = inter / (a1 + a2 - inter);
          if (iou > NMS_TH) sc[j] = -1.0f;
        }
      }
    }
    __syncthreads();
  }
}

// ---------------- host launcher ----------------
extern "C" void kernel_launch(void* const* d_in, const int* in_sizes, int n_in,
                              void* d_out, int out_size, void* d_ws, size_t ws_size,
                              hipStream_t stream) {
  const float* label_pre = (const float*)d_in[0]; // (B*N, 81)
  const float* bbox_pre  = (const float*)d_in[1]; // (B*N, 320)
  const float* proposals = (const float*)d_in[2]; // (B, N, 4)
  float* out = (float*)d_out;                     // 1600 + 400 + 400 floats

  // workspace layout (~56.4 MB): header | boxes | masked | cidx | cscore | cbox
  float* wsf = (float*)d_ws;
  unsigned* img_max_enc = (unsigned*)d_ws;                 // [BIMG]
  int* cnt = (int*)d_ws + BIMG;                            // [BIMG]
  float* boxes_ws  = wsf + 16;                             // TOTC*4
  float* masked_ws = boxes_ws + (size_t)TOTC * 4;          // TOTC
  int*   cidx      = (int*)(masked_ws + (size_t)TOTC);     // TOTC
  float* cscore    = (float*)(cidx + (size_t)TOTC);        // TOTC
  float* cbox      = cscore + (size_t)TOTC;                // TOTC*4

  init_kernel<<<1, 32, 0, stream>>>(img_max_enc, cnt);
  decode_kernel<<<NROWS / 16 / 8, 256, 0, stream>>>(label_pre, bbox_pre, proposals,
                                                    boxes_ws, masked_ws, img_max_enc);
  compact_kernel<<<640, 256, 0, stream>>>(masked_ws, boxes_ws, img_max_enc,
                                          cnt, cidx, cscore, cbox);
  nms_kernel<<<BIMG, 1024, 0, stream>>>(cnt, cidx, cscore, cbox, boxes_ws, out);
}